// SecondOrderOscillator_9096740733410
// MI455X (gfx1250) — compile-verified
//
#include <hip/hip_runtime.h>
#include <math.h>

typedef float v2f __attribute__((ext_vector_type(2)));
typedef float v8f __attribute__((ext_vector_type(8)));

#define CHUNK 256   // level-1 chunk length (inter-chunk scan granularity)
#define SUB   16    // WMMA tile length in time
#define NSUB  (CHUNK / SUB)

// ---------------- common device helpers ----------------

__device__ __forceinline__ void iir_coeffs(const float* __restrict__ rho,
                                           const float* __restrict__ psi,
                                           float& c1, float& c2) {
  // r = sigmoid(rho), theta = pi * sigmoid(psi)
  float r  = 1.0f / (1.0f + expf(-rho[0]));
  float th = 3.14159265358979323846f / (1.0f + expf(-psi[0]));
  c1 = 2.0f * r * cosf(th);   // = -a1
  c2 = -r * r;                // = -a2
}

__device__ __forceinline__ float load_u(const float* __restrict__ u,
                                        const float* __restrict__ u_init,
                                        int b, int t, int T, int nb) {
  if (t >= 0) return u[(size_t)b * T + t];
  return u_init[(size_t)b * (nb - 1) + (-t - 1)];   // u_in[-m] = u_init[b, m-1]
}

__device__ __forceinline__ float fir_v(const float* __restrict__ bc,
                                       const float* __restrict__ u,
                                       const float* __restrict__ u_init,
                                       int b, int t, int T, int nb) {
  float v = 0.0f;
  for (int k = 0; k < nb; ++k)
    v = fmaf(bc[k], load_u(u, u_init, b, t - k, T, nb), v);
  return v;
}

// ---------------- kernel 1: per-chunk particular finals ----------------
// One lane per (row, chunk): 256 zero-init recurrence steps over v(u).
// Templated on FIR order: coefficients in registers, float4 streaming loads,
// register sliding window for the taps.
template <int NB>
__global__ void __launch_bounds__(256)
k1_partials_t(const float* __restrict__ bc, const float* __restrict__ rho,
              const float* __restrict__ psi, const float* __restrict__ u,
              const float* __restrict__ u_init, float2* __restrict__ P,
              int B, int T, int C2) {
  int idx = blockIdx.x * blockDim.x + threadIdx.x;
  if (idx >= B * C2) return;
  int b = idx / C2, j = idx % C2;
  float c1, c2; iir_coeffs(rho, psi, c1, c2);

  float bcr[NB];
#pragma unroll
  for (int i = 0; i < NB; ++i) bcr[i] = bc[i];

  int t0 = j * CHUNK;
  float hist[NB > 1 ? NB - 1 : 1];   // hist[i] = u[t-1-i]
#pragma unroll
  for (int i = 0; i < NB - 1; ++i)
    hist[i] = load_u(u, u_init, b, t0 - 1 - i, T, NB);

  float y1 = 0.0f, y2 = 0.0f;
  const float4* up = (const float4*)(u + (size_t)b * T + t0);  // 16B aligned
  for (int t4 = 0; t4 < CHUNK / 4; ++t4) {
    float4 uu = up[t4];
    float uq[4] = {uu.x, uu.y, uu.z, uu.w};
#pragma unroll
    for (int q = 0; q < 4; ++q) {
      float v = bcr[0] * uq[q];
#pragma unroll
      for (int i = 1; i < NB; ++i) v = fmaf(bcr[i], hist[i - 1], v);
      float yy = fmaf(c1, y1, fmaf(c2, y2, v));
      y2 = y1; y1 = yy;
#pragma unroll
      for (int i = NB - 2; i > 0; --i) hist[i] = hist[i - 1];
      if (NB > 1) hist[0] = uq[q];
    }
  }
  P[idx] = make_float2(y1, y2);   // particular (y_{255}, y_{254})
}

// Generic (runtime nb) version.
__global__ void __launch_bounds__(256)
k1_partials_g(const float* __restrict__ bc, const float* __restrict__ rho,
              const float* __restrict__ psi, const float* __restrict__ u,
              const float* __restrict__ u_init, float2* __restrict__ P,
              int B, int T, int nb, int C2) {
  int idx = blockIdx.x * blockDim.x + threadIdx.x;
  if (idx >= B * C2) return;
  int b = idx / C2, j = idx % C2;
  float c1, c2; iir_coeffs(rho, psi, c1, c2);
  float y1 = 0.0f, y2 = 0.0f;
  int t0 = j * CHUNK;
  for (int t = 0; t < CHUNK; ++t) {
    float v = fir_v(bc, u, u_init, b, t0 + t, T, nb);
    float yy = fmaf(c1, y1, fmaf(c2, y2, v));
    y2 = y1; y1 = yy;
  }
  P[idx] = make_float2(y1, y2);
}

// ---------------- kernel 2: inter-chunk scan (per row) ----------------
// s_j = A^256 * s_{j-1} + p_j ; store the INCOMING state of each chunk.
__global__ void __launch_bounds__(256)
k2_scan(const float* __restrict__ rho, const float* __restrict__ psi,
        const float* __restrict__ y_init, const float2* __restrict__ P,
        float2* __restrict__ S, int B, int C2) {
  int b = blockIdx.x * blockDim.x + threadIdx.x;
  if (b >= B) return;
  float c1, c2; iir_coeffs(rho, psi, c1, c2);
  float h0 = c1, h1 = 1.0f, h2 = 0.0f;   // h_1, h_0, h_{-1}
  for (int t = 2; t <= CHUNK; ++t) {
    float hn = fmaf(c1, h0, c2 * h1);
    h2 = h1; h1 = h0; h0 = hn;
  }
  // A^256 = [[h256, c2*h255],[h255, c2*h254]]
  float m11 = h0, m12 = c2 * h1, m21 = h1, m22 = c2 * h2;
  float s1 = y_init[2 * b + 0], s2 = y_init[2 * b + 1];
  for (int j = 0; j < C2; ++j) {
    S[(size_t)b * C2 + j] = make_float2(s1, s2);
    float2 p = P[(size_t)b * C2 + j];
    float n1 = fmaf(m11, s1, fmaf(m12, s2, p.x));
    float n2 = fmaf(m21, s1, fmaf(m22, s2, p.y));
    s1 = n1; s2 = n2;
  }
}

// ---------------- kernel 3: WMMA Toeplitz solve + boundary fix --------
// One wave per (16-row group, 256-chunk). Per 16-step sub-block:
//   D = H * V  via 4x V_WMMA_F32_16X16X4_F32  (K accumulation)
//   y_tile = D + alpha*y_{-1} + beta*y_{-2};  state advanced with A^16.
// FIR==-1 selects the generic runtime-nb path.
template <int FIR>
__global__ void __launch_bounds__(256)
k3_wmma_t(const float* __restrict__ bc, const float* __restrict__ rho,
          const float* __restrict__ psi, const float* __restrict__ u,
          const float* __restrict__ u_init, const float2* __restrict__ S,
          float* __restrict__ y, int B, int T, int nb, int C2, int totalWaves) {
  __shared__ float h_s[SUB + 1];   // h_0 .. h_16 for dynamic H indexing

  float c1, c2; iir_coeffs(rho, psi, c1, c2);

  // local impulse response h_0..h_16 (compile-time indexed)
  float hh[SUB + 1];
  hh[0] = 1.0f; hh[1] = c1;
#pragma unroll
  for (int t = 2; t <= SUB; ++t) hh[t] = fmaf(c1, hh[t - 1], c2 * hh[t - 2]);

  if (threadIdx.x == 0) {
#pragma unroll
    for (int t = 0; t <= SUB; ++t) h_s[t] = hh[t];
  }
  __syncthreads();

  int wid = (blockIdx.x * blockDim.x + threadIdx.x) >> 5;
  if (wid >= totalWaves) return;
  int lane = threadIdx.x & 31;
  int half = lane >> 4;        // upper-half lanes hold K/M rows +2/+8 per ISA layout
  int n    = lane & 15;        // column (batch row within group); also A's M index

  int rg = wid / C2;           // 16-row group
  int j  = wid % C2;           // chunk index
  int b_row = rg * 16 + n;

  // FIR coefficients in registers (templated path)
  float bcr[FIR > 0 ? FIR : 1];
  if (FIR > 0) {
#pragma unroll
    for (int i = 0; i < (FIR > 0 ? FIR : 1); ++i) bcr[i] = bc[i];
  }

  // A-matrix (H, 16x16 lower-tri Toeplitz) in WMMA layout:
  // per WMMA kb: VGPR0 -> K = 4kb + 2*half, VGPR1 -> K+1; lane M = lane&15.
  v2f Ar[4];
#pragma unroll
  for (int kb = 0; kb < 4; ++kb) {
    int K0 = 4 * kb + 2 * half;
    Ar[kb][0] = (n >= K0)     ? h_s[n - K0]     : 0.0f;
    Ar[kb][1] = (n >= K0 + 1) ? h_s[n - K0 - 1] : 0.0f;
  }

  // boundary coefficients for this lane's D rows (M = i + 8*half):
  // alpha_m = h_{m+1}, beta_m = c2 * h_m
  float am[8], bm[8];
#pragma unroll
  for (int i = 0; i < 8; ++i) {
    am[i] = half ? hh[i + 9] : hh[i + 1];
    bm[i] = c2 * (half ? hh[i + 8] : hh[i]);
  }
  // A^16 for the intra-chunk state hop
  float m11 = hh[16], m12 = c2 * hh[15], m21 = hh[15], m22 = c2 * hh[14];

  float2 s = S[(size_t)b_row * C2 + j];   // incoming chunk state (y_{-1}, y_{-2})
  float s1 = s.x, s2 = s.y;

  const float* ubase = u + (size_t)b_row * T;

  for (int k = 0; k < NSUB; ++k) {
    int tb = j * CHUNK + k * SUB;

    // B-matrix (V = FIR(u), 16x16, K=time rows, N=batch cols), WMMA layout
    v2f Bv[4];
    if (FIR > 0 && tb >= 16) {
      // fast path: three aligned float2 loads give window u[tb+Kp-4 .. tb+Kp+1]
#pragma unroll
      for (int kb = 0; kb < 4; ++kb) {
        int Kp = 4 * kb + 2 * half;
        float2 wa = *(const float2*)(ubase + tb + Kp - 4);
        float2 wb = *(const float2*)(ubase + tb + Kp - 2);
        float2 wc = *(const float2*)(ubase + tb + Kp);
        float w[6] = {wa.x, wa.y, wb.x, wb.y, wc.x, wc.y};
        float v0 = 0.0f, v1 = 0.0f;
#pragma unroll
        for (int i = 0; i < (FIR > 0 ? FIR : 1); ++i) {
          v0 = fmaf(bcr[i], w[4 - i], v0);
          v1 = fmaf(bcr[i], w[5 - i], v1);
        }
        Bv[kb][0] = v0; Bv[kb][1] = v1;
      }
    } else if (FIR > 0) {
      // first tile of the sequence: guarded taps (rare, wave-uniform branch)
#pragma unroll
      for (int kb = 0; kb < 4; ++kb) {
        int Kp = 4 * kb + 2 * half;
        float v0 = 0.0f, v1 = 0.0f;
#pragma unroll
        for (int i = 0; i < (FIR > 0 ? FIR : 1); ++i) {
          v0 = fmaf(bcr[i], load_u(u, u_init, b_row, tb + Kp - i,     T, FIR), v0);
          v1 = fmaf(bcr[i], load_u(u, u_init, b_row, tb + Kp + 1 - i, T, FIR), v1);
        }
        Bv[kb][0] = v0; Bv[kb][1] = v1;
      }
    } else {
      // generic runtime-nb path
#pragma unroll
      for (int kb = 0; kb < 4; ++kb) {
        int Kp = 4 * kb + 2 * half;
        Bv[kb][0] = fir_v(bc, u, u_init, b_row, tb + Kp,     T, nb);
        Bv[kb][1] = fir_v(bc, u, u_init, b_row, tb + Kp + 1, T, nb);
      }
    }
    __builtin_prefetch(ubase + tb + SUB, 0, 1);

    // D = H * V  (f32 WMMA, K accumulated over 4 steps of 4)
    v8f D = {0.f, 0.f, 0.f, 0.f, 0.f, 0.f, 0.f, 0.f};
#pragma unroll
    for (int kb = 0; kb < 4; ++kb)
      D = __builtin_amdgcn_wmma_f32_16x16x4_f32(false, Ar[kb], false, Bv[kb],
                                                (short)0, D, false, false);

    // particular finals: rows M=15,14 live in upper-half lanes (VGPR 7,6);
    // broadcast to both halves with a cross-lane shuffle (ds_bpermute).
    float p1 = __shfl(D[7], 16 + n, 32);
    float p2 = __shfl(D[6], 16 + n, 32);

    // boundary correction: y = D + alpha*s1 + beta*s2
#pragma unroll
    for (int i = 0; i < 8; ++i) D[i] = fmaf(am[i], s1, fmaf(bm[i], s2, D[i]));

    // store 8 contiguous time samples per lane (32B aligned)
    float4* outp = (float4*)(y + (size_t)b_row * T + tb + 8 * half);
    outp[0] = make_float4(D[0], D[1], D[2], D[3]);
    outp[1] = make_float4(D[4], D[5], D[6], D[7]);

    // advance state across the sub-block: s_k = A^16 s_{k-1} + p_k
    float ns1 = fmaf(m11, s1, fmaf(m12, s2, p1));
    float ns2 = fmaf(m21, s1, fmaf(m22, s2, p2));
    s1 = ns1; s2 = ns2;
  }
}

// ---------------- fallback: plain serial per-row ----------------
__global__ void __launch_bounds__(256)
k_fallback(const float* __restrict__ bc, const float* __restrict__ rho,
           const float* __restrict__ psi, const float* __restrict__ u,
           const float* __restrict__ y_init, const float* __restrict__ u_init,
           float* __restrict__ y, int B, int T, int nb) {
  int b = blockIdx.x * blockDim.x + threadIdx.x;
  if (b >= B) return;
  float c1, c2; iir_coeffs(rho, psi, c1, c2);
  float y1 = y_init[2 * b + 0], y2 = y_init[2 * b + 1];
  for (int t = 0; t < T; ++t) {
    float v = fir_v(bc, u, u_init, b, t, T, nb);
    float yy = fmaf(c1, y1, fmaf(c2, y2, v));
    y[(size_t)b * T + t] = yy;
    y2 = y1; y1 = yy;
  }
}

// ---------------- launch ----------------
extern "C" void kernel_launch(void* const* d_in, const int* in_sizes, int n_in,
                              void* d_out, int out_size, void* d_ws, size_t ws_size,
                              hipStream_t stream) {
  const float* bc     = (const float*)d_in[0];
  const float* rho    = (const float*)d_in[1];
  const float* psi    = (const float*)d_in[2];
  const float* u      = (const float*)d_in[3];
  const float* y_init = (const float*)d_in[4];
  const float* u_init = (const float*)d_in[5];
  float* y = (float*)d_out;

  int nb = in_sizes[0];
  int B  = in_sizes[4] / 2;
  int T  = (B > 0) ? (in_sizes[3] / B) : 0;
  if (B <= 0 || T <= 0) return;

  int C2 = T / CHUNK;
  size_t need = (size_t)B * (size_t)(C2 > 0 ? C2 : 1) * sizeof(float2) * 2;

  if ((B % 16) != 0 || (T % CHUNK) != 0 || ws_size < need) {
    int blocks = (B + 255) / 256;
    k_fallback<<<blocks, 256, 0, stream>>>(bc, rho, psi, u, y_init, u_init, y, B, T, nb);
    return;
  }

  float2* P = (float2*)d_ws;
  float2* S = P + (size_t)B * C2;

  int g1 = (B * C2 + 255) / 256;
  int g2 = (B + 255) / 256;
  int totalWaves = (B / 16) * C2;
  int g3 = (totalWaves + 7) / 8;   // 8 waves (256 threads) per block

  switch (nb) {
    case 1:
      k1_partials_t<1><<<g1, 256, 0, stream>>>(bc, rho, psi, u, u_init, P, B, T, C2);
      k2_scan<<<g2, 256, 0, stream>>>(rho, psi, y_init, P, S, B, C2);
      k3_wmma_t<1><<<g3, 256, 0, stream>>>(bc, rho, psi, u, u_init, S, y, B, T, nb, C2, totalWaves);
      break;
    case 2:
      k1_partials_t<2><<<g1, 256, 0, stream>>>(bc, rho, psi, u, u_init, P, B, T, C2);
      k2_scan<<<g2, 256, 0, stream>>>(rho, psi, y_init, P, S, B, C2);
      k3_wmma_t<2><<<g3, 256, 0, stream>>>(bc, rho, psi, u, u_init, S, y, B, T, nb, C2, totalWaves);
      break;
    case 3:
      k1_partials_t<3><<<g1, 256, 0, stream>>>(bc, rho, psi, u, u_init, P, B, T, C2);
      k2_scan<<<g2, 256, 0, stream>>>(rho, psi, y_init, P, S, B, C2);
      k3_wmma_t<3><<<g3, 256, 0, stream>>>(bc, rho, psi, u, u_init, S, y, B, T, nb, C2, totalWaves);
      break;
    case 4:
      k1_partials_t<4><<<g1, 256, 0, stream>>>(bc, rho, psi, u, u_init, P, B, T, C2);
      k2_scan<<<g2, 256, 0, stream>>>(rho, psi, y_init, P, S, B, C2);
      k3_wmma_t<4><<<g3, 256, 0, stream>>>(bc, rho, psi, u, u_init, S, y, B, T, nb, C2, totalWaves);
      break;
    case 5:
      k1_partials_t<5><<<g1, 256, 0, stream>>>(bc, rho, psi, u, u_init, P, B, T, C2);
      k2_scan<<<g2, 256, 0, stream>>>(rho, psi, y_init, P, S, B, C2);
      k3_wmma_t<5><<<g3, 256, 0, stream>>>(bc, rho, psi, u, u_init, S, y, B, T, nb, C2, totalWaves);
      break;
    default:
      k1_partials_g<<<g1, 256, 0, stream>>>(bc, rho, psi, u, u_init, P, B, T, nb, C2);
      k2_scan<<<g2, 256, 0, stream>>>(rho, psi, y_init, P, S, B, C2);
      k3_wmma_t<-1><<<g3, 256, 0, stream>>>(bc, rho, psi, u, u_init, S, y, B, T, nb, C2, totalWaves);
      break;
  }
}